// UniGCNIILayer_2576980378136
// MI455X (gfx1250) — compile-verified
//
#include <hip/hip_runtime.h>
#include <cstddef>

typedef __attribute__((ext_vector_type(2))) float v2f;
typedef __attribute__((ext_vector_type(8))) float v8f;

#define FEAT_D 128   // feature dim, fixed by the model

// ---------------------------------------------------------------------------
// zero: float4-vectorized memset(0)
// ---------------------------------------------------------------------------
__global__ void zero_f32(float4* __restrict__ p, int n4) {
  int i = blockIdx.x * blockDim.x + threadIdx.x;
  if (i < n4) p[i] = make_float4(0.f, 0.f, 0.f, 0.f);
}

// ---------------------------------------------------------------------------
// Stage 1: one wave32 per nnz entry. 32 lanes x float4 == one 128-f32 row.
// eacc[e] += vfeat[n];  ecnt[e] += 1
// ---------------------------------------------------------------------------
__global__ void scatter_v2e(const float* __restrict__ vfeat,
                            const int* __restrict__ node_idx,
                            const int* __restrict__ edge_idx,
                            float* __restrict__ eacc,
                            float* __restrict__ ecnt, int nnz) {
  int gtid = blockIdx.x * blockDim.x + threadIdx.x;
  int wid  = gtid >> 5;
  int lane = threadIdx.x & 31;
  if (wid >= nnz) return;
  int n = node_idx[wid];
  int e = edge_idx[wid];
  float4 val = reinterpret_cast<const float4*>(vfeat + (size_t)n * FEAT_D)[lane];
  float* dst = eacc + (size_t)e * FEAT_D + lane * 4;
  atomicAdd(dst + 0, val.x);
  atomicAdd(dst + 1, val.y);
  atomicAdd(dst + 2, val.z);
  atomicAdd(dst + 3, val.w);
  if (lane == 0) atomicAdd(ecnt + e, 1.0f);
}

// efeat_new = eacc / max(cnt, 1)   (in place, E*D elements)
__global__ void normalize_e(float* __restrict__ eacc,
                            const float* __restrict__ ecnt, int total) {
  int i = blockIdx.x * blockDim.x + threadIdx.x;
  if (i >= total) return;
  float c = ecnt[i >> 7];                 // i / 128
  eacc[i] = eacc[i] / fmaxf(c, 1.0f);
}

// ---------------------------------------------------------------------------
// Stage 2: one wave32 per nnz entry.
// hacc[n] += degE[e] * efeat_new[e]   (efeat_new is L2-resident: 10 MB)
// ---------------------------------------------------------------------------
__global__ void scatter_e2v(const float* __restrict__ efeat_new,
                            const float* __restrict__ degE,
                            const int* __restrict__ node_idx,
                            const int* __restrict__ edge_idx,
                            float* __restrict__ hacc, int nnz) {
  int gtid = blockIdx.x * blockDim.x + threadIdx.x;
  int wid  = gtid >> 5;
  int lane = threadIdx.x & 31;
  if (wid >= nnz) return;
  int n = node_idx[wid];
  int e = edge_idx[wid];
  float w = degE[e];
  float4 val = reinterpret_cast<const float4*>(efeat_new + (size_t)e * FEAT_D)[lane];
  float* dst = hacc + (size_t)n * FEAT_D + lane * 4;
  atomicAdd(dst + 0, w * val.x);
  atomicAdd(dst + 1, w * val.y);
  atomicAdd(dst + 2, w * val.z);
  atomicAdd(dst + 3, w * val.w);
}

// vi = (1-a) * degV[row] * hacc + a * vfeat0     (in place over hacc)
__global__ void make_vi(float* __restrict__ hacc,
                        const float* __restrict__ degV,
                        const float* __restrict__ vfeat0,
                        const float* __restrict__ alpha, int total) {
  int i = blockIdx.x * blockDim.x + threadIdx.x;
  if (i >= total) return;
  float a = alpha[0];
  float h = hacc[i] * degV[i >> 7];
  hacc[i] = (1.0f - a) * h + a * vfeat0[i];
}

// ---------------------------------------------------------------------------
// Stage 3: v = (1-b)*vi + b*(vi @ W^T)  with fp32 WMMA (V_WMMA_F32_16X16X4_F32)
// Block = 256 threads = 8 waves; W staged in 64 KB LDS; each wave owns one
// 16-row M-tile and all 8 column tiles (8 x v8f accumulators).
// ---------------------------------------------------------------------------
__global__ void gemm_wmma(const float* __restrict__ vi,
                          const float* __restrict__ W,
                          const float* __restrict__ beta,
                          float* __restrict__ out, int nrows) {
  __shared__ float Wlds[FEAT_D * FEAT_D];   // 64 KB, row-major W[j][k]

  const int tid = threadIdx.x;
  // cooperative W load: 4096 float4 / 256 threads = 16 each
  for (int t = tid; t < (FEAT_D * FEAT_D) / 4; t += blockDim.x) {
    reinterpret_cast<float4*>(Wlds)[t] = reinterpret_cast<const float4*>(W)[t];
  }
  __syncthreads();

  const float b   = beta[0];
  const float omb = 1.0f - b;
  const int wave  = tid >> 5;       // 0..7
  const int lane  = tid & 31;
  const int half  = lane >> 4;      // 0 | 1
  const int r     = lane & 15;      // 0..15
  const int m0    = (blockIdx.x * 8 + wave) * 16;
  if (m0 >= nrows) return;

  v8f acc[8] = {};                  // 8 col-tiles x 16x16 f32 accum

  const float* arow = vi + (size_t)(m0 + r) * FEAT_D;
  for (int k0 = 0; k0 < FEAT_D; k0 += 4) {
    // A 16x4 fragment: lane(half,r) holds A[m0+r][k0+2*half + {0,1}]
    v2f a = *reinterpret_cast<const v2f*>(arow + k0 + 2 * half);
#pragma unroll
    for (int j = 0; j < 8; ++j) {
      const int n0 = j * 16;
      // B 4x16 fragment: B[k][n] = W[n][k] -> lane holds W[n0+r][k0+2*half + {0,1}]
      v2f bf = *reinterpret_cast<const v2f*>(&Wlds[(n0 + r) * FEAT_D + k0 + 2 * half]);
      acc[j] = __builtin_amdgcn_wmma_f32_16x16x4_f32(
          /*neg_a=*/false, a, /*neg_b=*/false, bf,
          /*c_mod=*/(short)0, acc[j], /*reuse_a=*/false, /*reuse_b=*/false);
    }
  }

  // Epilogue: C/D layout -> VGPR g = row m0+g+8*half, col n0+r
#pragma unroll
  for (int j = 0; j < 8; ++j) {
    const int n0 = j * 16;
#pragma unroll
    for (int g = 0; g < 8; ++g) {
      const size_t idx = (size_t)(m0 + g + 8 * half) * FEAT_D + (n0 + r);
      out[idx] = omb * vi[idx] + b * acc[j][g];
    }
  }
}

// ---------------------------------------------------------------------------
extern "C" void kernel_launch(void* const* d_in, const int* in_sizes, int n_in,
                              void* d_out, int out_size, void* d_ws, size_t ws_size,
                              hipStream_t stream) {
  const float* vfeat   = (const float*)d_in[0];
  const float* degE    = (const float*)d_in[2];
  const float* degV    = (const float*)d_in[3];
  const float* vfeat0  = (const float*)d_in[4];
  const float* W       = (const float*)d_in[5];
  const float* alpha   = (const float*)d_in[6];
  const float* beta    = (const float*)d_in[7];
  const int* node_idx  = (const int*)d_in[8];
  const int* edge_idx  = (const int*)d_in[9];

  const int E   = in_sizes[2];   // degE
  const int N   = in_sizes[3];   // degV
  const int NNZ = in_sizes[8];   // node_idx

  // d_out = [ v : N*D | efeat_new : E*D ]
  float* out_v     = (float*)d_out;
  float* efeat_new = out_v + (size_t)N * FEAT_D;   // accumulate + normalize in place

  // d_ws = [ ecnt : E | hacc/vi : N*D ]
  float* ecnt = (float*)d_ws;
  float* hacc = ecnt + E;

  const int B = 256;

  // zero accumulators
  {
    int n4 = (E + N * FEAT_D) / 4;
    zero_f32<<<(n4 + B - 1) / B, B, 0, stream>>>((float4*)d_ws, n4);
    int m4 = (E * FEAT_D) / 4;
    zero_f32<<<(m4 + B - 1) / B, B, 0, stream>>>((float4*)efeat_new, m4);
  }

  // Stage 1: node -> edge mean
  {
    long threads = (long)NNZ * 32;
    scatter_v2e<<<(int)((threads + B - 1) / B), B, 0, stream>>>(
        vfeat, node_idx, edge_idx, efeat_new, ecnt, NNZ);
    int tot = E * FEAT_D;
    normalize_e<<<(tot + B - 1) / B, B, 0, stream>>>(efeat_new, ecnt, tot);
  }

  // Stage 2: edge -> node weighted sum, then vi
  {
    long threads = (long)NNZ * 32;
    scatter_e2v<<<(int)((threads + B - 1) / B), B, 0, stream>>>(
        efeat_new, degE, node_idx, edge_idx, hacc, NNZ);
    int tot = N * FEAT_D;
    make_vi<<<(tot + B - 1) / B, B, 0, stream>>>(hacc, degV, vfeat0, alpha, tot);
  }

  // Stage 3: fused GCNII GEMM with fp32 WMMA
  {
    int mtiles = (N + 15) / 16;
    int blocks = (mtiles + 7) / 8;
    gemm_wmma<<<blocks, 256, 0, stream>>>(hacc, W, beta, out_v, N);
  }
}